// BodyModelTorch_57913339019685
// MI455X (gfx1250) — compile-verified
//
#include <hip/hip_runtime.h>
#include <hip/hip_bf16.h>

#define BATCH 256
#define KJ    62
#define NVERT 14522
#define NBONE 28
#define KPAD  64

typedef float v2f __attribute__((ext_vector_type(2)));
typedef float v8f __attribute__((ext_vector_type(8)));

// ---------------------------------------------------------------------------
// Kernel 1: per-batch prep. One block (64 threads) per batch element.
//  - bone-length factors, Euler->R, A matrices (in LDS)
//  - sequential kinematic chain (12 threads compute a 3x4 G row-block per step)
//  - J_out = G[:, :3, 3]*scale + trans
//  - G translation fixup (t' = t - R_global @ Jt)
//  - emit padded column-major WMMA B operand: gbt[b][col(16)][krow(64)]
// ---------------------------------------------------------------------------
__global__ __launch_bounds__(64) void prep_kernel(
    const float* __restrict__ thetas,   // [B,186] (z,y,x per joint)
    const float* __restrict__ blcore,   // [B,28]
    const float* __restrict__ cbl,      // [B,1]
    const float* __restrict__ trans,    // [B,3]
    const float* __restrict__ scale,    // [B]
    const float* __restrict__ tpj,      // [62,3]
    const int*   __restrict__ parents,  // [62]
    const int*   __restrict__ mapper,   // [62]
    float* __restrict__ gbt,            // [B,16,64] col-major padded G
    float* __restrict__ jout)           // [B,62,3]
{
  __shared__ float A[KJ][12];
  __shared__ float G[KJ][12];
  __shared__ int   par[KJ];
  const int b = blockIdx.x;
  const int t = threadIdx.x;

  if (t < KJ) {
    const int k  = t;
    const int p  = parents[k];
    par[k] = p;
    const int   mp     = mapper[k];
    const int   mc     = (mp < 0) ? 0 : mp;
    const float core   = blcore[b * NBONE + mc];
    const float blc    = 2.0f / (1.0f + __expf(-core * 0.2f)); // sigmoid(x/5)*2
    float factor = (mp < 0) ? 1.0f : blc;
    if (k == 1) factor = cbl[b];
    if (k == 0) factor = 1.0f;

    const float jkx = tpj[k*3+0], jky = tpj[k*3+1], jkz = tpj[k*3+2];
    const float jpx = tpj[p*3+0], jpy = tpj[p*3+1], jpz = tpj[p*3+2];
    float ox, oy, oz;
    if (k == 0) { ox = jkx; oy = jky; oz = jkz; }
    else {
      ox = (jkx - jpx) * factor;
      oy = (jky - jpy) * factor;
      oz = (jkz - jpz) * factor;
    }

    const float az = thetas[b*186 + 3*k + 0];  // z
    const float ay = thetas[b*186 + 3*k + 1];  // y
    const float ax = thetas[b*186 + 3*k + 2];  // x
    const float cx = cosf(ax), sx = sinf(ax);
    const float cy = cosf(ay), sy = sinf(ay);
    const float cz = cosf(az), sz = sinf(az);
    // R = Rz @ Ry @ Rx
    A[k][0]  = cz*cy;               A[k][1]  = cz*sy*sx - sz*cx;
    A[k][2]  = cz*sy*cx + sz*sx;    A[k][3]  = ox;
    A[k][4]  = sz*cy;               A[k][5]  = sz*sy*sx + cz*cx;
    A[k][6]  = sz*sy*cx - cz*sx;    A[k][7]  = oy;
    A[k][8]  = -sy;                 A[k][9]  = cy*sx;
    A[k][10] = cy*cx;               A[k][11] = oz;
  }
  __syncthreads();

  if (t < 12) G[0][t] = A[0][t];
  __syncthreads();

  for (int k = 1; k < KJ; ++k) {
    if (t < 12) {
      const int i = t >> 2, j = t & 3;
      const int p = par[k];
      float acc = (j == 3) ? G[p][i*4+3] : 0.0f;
      acc += G[p][i*4+0] * A[k][0*4+j];
      acc += G[p][i*4+1] * A[k][1*4+j];
      acc += G[p][i*4+2] * A[k][2*4+j];
      G[k][t] = acc;
    }
    __syncthreads();
  }

  if (t < KJ) {
    const int k = t;
    const float s  = scale[b];
    const float tx = trans[b*3+0], ty = trans[b*3+1], tz = trans[b*3+2];
    jout[((size_t)b*KJ + k)*3 + 0] = G[k][3]  * s + tx;
    jout[((size_t)b*KJ + k)*3 + 1] = G[k][7]  * s + ty;
    jout[((size_t)b*KJ + k)*3 + 2] = G[k][11] * s + tz;
    // translation fixup: t' = t - R_global @ Jt[k]
    const float jx = tpj[k*3+0], jy = tpj[k*3+1], jz = tpj[k*3+2];
    G[k][3]  -= G[k][0]*jx + G[k][1]*jy + G[k][2]*jz;
    G[k][7]  -= G[k][4]*jx + G[k][5]*jy + G[k][6]*jz;
    G[k][11] -= G[k][8]*jx + G[k][9]*jy + G[k][10]*jz;
  }
  __syncthreads();

  // Emit zero-padded col-major B operand: gbt[b][col][krow], krow = t (0..63)
  {
    float* dst = gbt + (size_t)b * 1024 + t;
    #pragma unroll
    for (int col = 0; col < 16; ++col) {
      float v = 0.0f;
      if (t < KJ && col < 12) v = G[t][col];
      dst[col * 64] = v;
    }
  }
}

// ---------------------------------------------------------------------------
// Kernel 2: per-batch GEMM  M[V x 12] = W[V x 62] @ Gflat[62 x 12]
// via V_WMMA_F32_16X16X4_F32 (K padded to 64, N padded to 16),
// fused with per-vertex homogeneous transform + scale/trans epilogue.
// One 16-row tile per wave; 8 waves (256 threads) per block; grid (114, B).
// ---------------------------------------------------------------------------
__global__ __launch_bounds__(256) void lbs_wmma_kernel(
    const float* __restrict__ weights,  // [V,62]
    const float* __restrict__ vtemp,    // [V,3]
    const float* __restrict__ scale,    // [B]
    const float* __restrict__ trans,    // [B,3]
    const float* __restrict__ gbt,      // [B,16,64]
    float* __restrict__ vout)           // [B,V,3]
{
  __shared__ float gb[1024];        // B operand for this batch (col-major)
  __shared__ float dtile[8][256];   // per-wave 16x16 D tiles

  const int b = blockIdx.y;
  for (int i = threadIdx.x; i < 1024; i += 256)
    gb[i] = gbt[(size_t)b * 1024 + i];
  __syncthreads();

  const int wave = threadIdx.x >> 5;
  const int lane = threadIdx.x & 31;
  const int half = lane >> 4;       // 0: lanes 0-15, 1: lanes 16-31
  const int col  = lane & 15;
  const int rowbase = (blockIdx.x * 8 + wave) * 16;
  const bool active = rowbase < NVERT;   // wave-uniform

  if (active) {
    const int arow = min(rowbase + col, NVERT - 1);     // clamp tail rows
    const float* wrow = weights + (size_t)arow * KJ;
    v8f c = {0.f, 0.f, 0.f, 0.f, 0.f, 0.f, 0.f, 0.f};
    #pragma unroll
    for (int kk = 0; kk < KPAD; kk += 4) {
      // A (16x4 f32): lanes 0-15 hold K=kk,kk+1; lanes 16-31 hold K=kk+2,kk+3
      const int kc = kk + 2 * half;
      v2f a = {0.0f, 0.0f};
      if (kc < KJ) a = *(const v2f*)(wrow + kc);        // 8B-aligned (kc, row*62 even)
      // B (4x16 f32): same K split across lane halves, N = col
      const v2f bm = *(const v2f*)(gb + col * 64 + kc);
      c = __builtin_amdgcn_wmma_f32_16x16x4_f32(
              /*neg_a=*/false, a, /*neg_b=*/false, bm,
              /*c_mod=*/(short)0, c, /*reuse_a=*/false, /*reuse_b=*/false);
    }
    // D layout: VGPR r, lane -> element (M = r + 8*half, N = col)
    #pragma unroll
    for (int r = 0; r < 8; ++r)
      dtile[wave][(r + 8 * half) * 16 + col] = c[r];
  }
  __syncthreads();

  if (active && lane < 16) {
    const int gr = rowbase + lane;
    if (gr < NVERT) {
      const float* M = &dtile[wave][lane * 16];
      const float x = vtemp[gr*3+0], y = vtemp[gr*3+1], z = vtemp[gr*3+2];
      const float s = scale[b];
      const float vx = M[0]*x + M[1]*y + M[2]*z  + M[3];
      const float vy = M[4]*x + M[5]*y + M[6]*z  + M[7];
      const float vz = M[8]*x + M[9]*y + M[10]*z + M[11];
      float* o = vout + ((size_t)b * NVERT + gr) * 3;
      o[0] = vx * s + trans[b*3+0];
      o[1] = vy * s + trans[b*3+1];
      o[2] = vz * s + trans[b*3+2];
    }
  }
}

// ---------------------------------------------------------------------------
extern "C" void kernel_launch(void* const* d_in, const int* in_sizes, int n_in,
                              void* d_out, int out_size, void* d_ws, size_t ws_size,
                              hipStream_t stream) {
  const float* thetas  = (const float*)d_in[0];
  const float* blcore  = (const float*)d_in[1];
  const float* cbl     = (const float*)d_in[2];
  const float* trans   = (const float*)d_in[3];
  const float* scale   = (const float*)d_in[4];
  const float* vtemp   = (const float*)d_in[5];
  const float* tpj     = (const float*)d_in[6];
  const float* weights = (const float*)d_in[7];
  const int*   parents = (const int*)d_in[8];
  const int*   mapper  = (const int*)d_in[9];

  float* vout = (float*)d_out;                          // [B,V,3]
  float* jout = vout + (size_t)BATCH * NVERT * 3;       // [B,K,3]
  float* gbt  = (float*)d_ws;                           // B*1024 floats = 1 MB

  prep_kernel<<<BATCH, 64, 0, stream>>>(thetas, blcore, cbl, trans, scale, tpj,
                                        parents, mapper, gbt, jout);

  const int tilesPerBatch  = (NVERT + 15) / 16;         // 908
  const int blocksPerBatch = (tilesPerBatch + 7) / 8;   // 114
  dim3 grid(blocksPerBatch, BATCH);
  lbs_wmma_kernel<<<grid, 256, 0, stream>>>(weights, vtemp, scale, trans, gbt, vout);
}